// DKVMN_58944131170332
// MI455X (gfx1250) — compile-verified
//
#include <hip/hip_runtime.h>
#include <math.h>

typedef float v2f __attribute__((ext_vector_type(2)));
typedef float v4f __attribute__((ext_vector_type(4)));
typedef float v8f __attribute__((ext_vector_type(8)));

#define B_   64
#define T_   200
#define NC   2000
#define DIMS 128
#define SM   50
#define SMP  64                 // Mk padded rows (WMMA shaped)
#define ROWS (B_ * T_)          // 12800
#define NTIL (ROWS / 16)        // 800 row-tiles

// LDS pitches chosen so lane banks (pitch%64)*c + 2*sel are distinct mod 64
#define PIT_A 132               // 132%64=4  -> banks 4c (+2 for hi half)
#define PIT_C 268               // 268%64=12 -> 12c distinct for c<16
#define PIT_W 68

__device__ __forceinline__ float sig_(float x) { return 1.0f / (1.0f + __expf(-x)); }

// f32 WMMA: D(16x16) = A(16x4) * B(4x16) + C ; exact fp32 accumulate
__device__ __forceinline__ v8f wmma_f32(v2f a, v2f b, v8f c) {
    return __builtin_amdgcn_wmma_f32_16x16x4_f32(false, a, false, b, (short)0, c,
                                                 false, false);
}

// ---------------- Kernel P: pad Mk (50x128) -> (64x128) zero-filled ---------
__global__ __launch_bounds__(256) void prep_mk(const float* __restrict__ Mk,
                                               float* __restrict__ mkp) {
    int i = blockIdx.x * 256 + threadIdx.x;          // 0..8191
    int m = i >> 7, kk = i & 127;
    mkp[i] = (m < SM) ? Mk[m * DIMS + kk] : 0.0f;
}

// ---------------- Kernel A: gather + e/a GEMMs + w GEMM + softmax -----------
__global__ __launch_bounds__(256) void eaw_kernel(
    const int* __restrict__ skill, const int* __restrict__ answer,
    const float* __restrict__ k_emb, const float* __restrict__ v_emb,
    const float* __restrict__ mkp,
    const float* __restrict__ e_W, const float* __restrict__ e_b,
    const float* __restrict__ a_W, const float* __restrict__ a_b,
    float* __restrict__ e_out, float* __restrict__ a_out,
    float* __restrict__ w_out) {
    __shared__ __align__(16) float lk[8][16][PIT_A];
    __shared__ __align__(16) float lv[8][16][PIT_A];
    __shared__ __align__(16) float lw[8][16][PIT_W];

    const int wave = threadIdx.x >> 5, lane = threadIdx.x & 31;
    const int c = lane & 15, sel = lane >> 4;
    const int R0 = (blockIdx.x * 8 + wave) * 16;

    // stage 16 gathered k/v rows: one 512B coalesced row per iteration
    for (int j = 0; j < 16; ++j) {
        int row = R0 + j;
        int sk = skill[row];
        int an = answer[row]; an = (an == 2) ? 1 : an;
        int xv = sk + NC * an;
        v4f kq = *(const v4f*)(k_emb + (size_t)sk * DIMS + lane * 4);
        v4f vq = *(const v4f*)(v_emb + (size_t)xv * DIMS + lane * 4);
        *(v4f*)&lk[wave][j][lane * 4] = kq;
        *(v4f*)&lv[wave][j][lane * 4] = vq;
    }
    __syncthreads();

    // ---- e = sigmoid(v @ e_W^T + b), a = tanh(v @ a_W^T + b); 8 N-tiles
    for (int jt = 0; jt < 8; ++jt) {
        v8f ae = {}, aa = {};
        const int N0 = jt * 16;
        for (int s = 0; s < 32; ++s) {
            int kk = 4 * s + 2 * sel;
            v2f af = *(const v2f*)&lv[wave][c][kk];                       // A: M=c
            v2f be = *(const v2f*)(e_W + (size_t)(N0 + c) * DIMS + kk);   // B: N=c
            v2f ba = *(const v2f*)(a_W + (size_t)(N0 + c) * DIMS + kk);
            ae = wmma_f32(af, be, ae);
            aa = wmma_f32(af, ba, aa);
        }
        float eb = e_b[N0 + c], ab = a_b[N0 + c];
#pragma unroll
        for (int r = 0; r < 8; ++r) {
            size_t row = (size_t)(R0 + 8 * sel + r);
            e_out[row * DIMS + N0 + c] = sig_(ae[r] + eb);
            a_out[row * DIMS + N0 + c] = tanhf(aa[r] + ab);
        }
    }

    // ---- w logits = k @ Mk^T (padded to 64 cols); 4 N-tiles -> LDS
    for (int jt = 0; jt < 4; ++jt) {
        v8f ac = {};
        const int N0 = jt * 16;
        for (int s = 0; s < 32; ++s) {
            int kk = 4 * s + 2 * sel;
            v2f af = *(const v2f*)&lk[wave][c][kk];
            v2f bm = *(const v2f*)(mkp + (size_t)(N0 + c) * DIMS + kk);
            ac = wmma_f32(af, bm, ac);
        }
#pragma unroll
        for (int r = 0; r < 8; ++r) lw[wave][8 * sel + r][N0 + c] = ac[r];
    }
    __syncthreads();

    // ---- softmax over m<50 : lane pair (c, c+16) owns row c, 32 cols each
    {
        const int mb = sel * 32;
        float vals[32];
        float mx = -1e30f;
#pragma unroll
        for (int m = 0; m < 32; ++m) {
            int mm = mb + m;
            float x = (mm < SM) ? lw[wave][c][mm] : -1e30f;
            vals[m] = x;
            mx = fmaxf(mx, x);
        }
        mx = fmaxf(mx, __shfl_xor(mx, 16, 32));
        float sum = 0.0f;
#pragma unroll
        for (int m = 0; m < 32; ++m) {
            float ex = (vals[m] > -1e29f) ? __expf(vals[m] - mx) : 0.0f;
            vals[m] = ex;
            sum += ex;
        }
        sum += __shfl_xor(sum, 16, 32);
        float inv = 1.0f / sum;
        size_t row = (size_t)(R0 + c);
#pragma unroll
        for (int m = 0; m < 32; ++m) w_out[row * SMP + mb + m] = vals[m] * inv;
    }
}

// ---------------- Kernel B: sequential scan, Mv in registers ---------------
__global__ __launch_bounds__(128) void scan_kernel(
    const float* __restrict__ Mv0, const float* __restrict__ e_in,
    const float* __restrict__ a_in, const float* __restrict__ w_in,
    float* __restrict__ reads) {
    __shared__ float wbuf[2][SMP];
    const int b = blockIdx.x;
    const int d = threadIdx.x;          // column 0..127
    float Mv[SM];
#pragma unroll
    for (int m = 0; m < SM; ++m) Mv[m] = Mv0[m * DIMS + d];

    for (int t = 0; t < T_; ++t) {
        const size_t row = (size_t)b * T_ + t;
        if (threadIdx.x < SMP) wbuf[t & 1][threadIdx.x] = w_in[row * SMP + threadIdx.x];
        float ed = e_in[row * DIMS + d];
        float ad = a_in[row * DIMS + d];
        __syncthreads();
        const float* wv = wbuf[t & 1];
        float rd = 0.0f;
#pragma unroll
        for (int m = 0; m < SM; ++m) {
            float wm = wv[m];                          // LDS broadcast
            rd = fmaf(wm, Mv[m], rd);                  // read uses OLD Mv
            Mv[m] = fmaf(wm, fmaf(-ed, Mv[m], ad), Mv[m]);
        }
        reads[row * DIMS + d] = rd;
    }
}

// -------- Kernel C: f = tanh([reads|k] @ f_W^T + b) ; gathered pred dot ----
__global__ __launch_bounds__(128) void out_kernel(
    const int* __restrict__ skill, const float* __restrict__ k_emb,
    const float* __restrict__ reads,
    const float* __restrict__ f_W, const float* __restrict__ f_b,
    const float* __restrict__ p_W, const float* __restrict__ p_b,
    float* __restrict__ pred) {
    __shared__ __align__(16) float lA[4][16][PIT_C];   // concat rows, K=256
    __shared__ __align__(16) float lF[4][16][PIT_A];   // f tile
    const int wave = threadIdx.x >> 5, lane = threadIdx.x & 31;
    const int c = lane & 15, sel = lane >> 4;
    const int R0 = (blockIdx.x * 4 + wave) * 16;

    for (int j = 0; j < 16; ++j) {
        int row = R0 + j;
        int sk = skill[row];
        v4f rv = *(const v4f*)(reads + (size_t)row * DIMS + lane * 4);
        v4f kv = *(const v4f*)(k_emb + (size_t)sk * DIMS + lane * 4);
        *(v4f*)&lA[wave][j][lane * 4] = rv;
        *(v4f*)&lA[wave][j][DIMS + lane * 4] = kv;
    }
    __syncthreads();

    for (int jt = 0; jt < 8; ++jt) {
        v8f ac = {};
        const int N0 = jt * 16;
        for (int s = 0; s < 64; ++s) {                 // K = 256
            int kk = 4 * s + 2 * sel;
            v2f af = *(const v2f*)&lA[wave][c][kk];
            v2f bf = *(const v2f*)(f_W + (size_t)(N0 + c) * (2 * DIMS) + kk);
            ac = wmma_f32(af, bf, ac);
        }
        float fb = f_b[N0 + c];
#pragma unroll
        for (int r = 0; r < 8; ++r)
            lF[wave][8 * sel + r][N0 + c] = tanhf(ac[r] + fb);
    }
    __syncthreads();

    // pred[b,t] = sigmoid(f[b,t] . p_W[skill[b,t+1]] + p_b[idx]); skip t==T-1
    for (int j = 0; j < 16; ++j) {
        int row = R0 + j;
        int t = row % T_;                  // uniform across the wave
        if (t == T_ - 1) continue;
        int idx = skill[row + 1];
        v4f fv = *(const v4f*)&lF[wave][j][lane * 4];
        v4f pv = *(const v4f*)(p_W + (size_t)idx * DIMS + lane * 4);
        float part = fv.x * pv.x + fv.y * pv.y + fv.z * pv.z + fv.w * pv.w;
#pragma unroll
        for (int off = 16; off >= 1; off >>= 1) part += __shfl_xor(part, off, 32);
        if (lane == 0) {
            int b = row / T_;
            pred[(size_t)b * (T_ - 1) + t] = sig_(part + p_b[idx]);
        }
    }
}

// ---------------------------------------------------------------------------
extern "C" void kernel_launch(void* const* d_in, const int* in_sizes, int n_in,
                              void* d_out, int out_size, void* d_ws, size_t ws_size,
                              hipStream_t stream) {
    const int*   skill  = (const int*)d_in[0];
    const int*   answer = (const int*)d_in[1];
    const float* k_emb  = (const float*)d_in[2];
    const float* v_emb  = (const float*)d_in[3];
    const float* Mk     = (const float*)d_in[4];
    const float* Mv0    = (const float*)d_in[5];
    const float* f_W    = (const float*)d_in[6];
    const float* f_b    = (const float*)d_in[7];
    const float* p_W    = (const float*)d_in[8];
    const float* p_b    = (const float*)d_in[9];
    const float* e_W    = (const float*)d_in[10];
    const float* e_b    = (const float*)d_in[11];
    const float* a_W    = (const float*)d_in[12];
    const float* a_b    = (const float*)d_in[13];
    float* pred = (float*)d_out;

    float* ws = (float*)d_ws;
    float* mkp     = ws;                       // 8192
    float* e_buf   = mkp + (size_t)SMP * DIMS; // 12800*128
    float* a_buf   = e_buf + (size_t)ROWS * DIMS;
    float* w_buf   = a_buf + (size_t)ROWS * DIMS;   // 12800*64
    float* r_buf   = w_buf + (size_t)ROWS * SMP;    // 12800*128

    prep_mk<<<(SMP * DIMS) / 256, 256, 0, stream>>>(Mk, mkp);
    eaw_kernel<<<NTIL / 8, 256, 0, stream>>>(skill, answer, k_emb, v_emb, mkp,
                                             e_W, e_b, a_W, a_b,
                                             e_buf, a_buf, w_buf);
    scan_kernel<<<B_, 128, 0, stream>>>(Mv0, e_buf, a_buf, w_buf, r_buf);
    out_kernel<<<NTIL / 4, 128, 0, stream>>>(skill, k_emb, r_buf, f_W, f_b,
                                             p_W, p_b, pred);
}